// GANLoss_56298431316074
// MI455X (gfx1250) — compile-verified
//
#include <hip/hip_runtime.h>
#include <hip/hip_bf16.h>

// Native clang vector types (work with __builtin_nontemporal_load and WMMA builtins)
typedef __attribute__((ext_vector_type(4))) float v4f;
typedef __attribute__((ext_vector_type(4))) int   v4i;
typedef __attribute__((ext_vector_type(2))) float v2f;
typedef __attribute__((ext_vector_type(8))) float v8f;

#define NB  4096   // partial-sum blocks (pass 1 grid)
#define NT_ 256    // threads per block (8 wave32 waves)

// ---------------------------------------------------------------------------
// Pass 1: streaming hinge-loss contribution + per-block reduction.
// Bandwidth-bound: 128-bit non-temporal loads, grid-stride, 32K waves in
// flight to saturate 23.3 TB/s HBM.
// ---------------------------------------------------------------------------
__global__ __launch_bounds__(NT_) void gan_hinge_partial(
    const float* __restrict__ D, const int* __restrict__ T,
    float* __restrict__ partials, int n)
{
    __shared__ float sdata[NT_];
    const int  tid    = threadIdx.x;
    const long gid    = (long)blockIdx.x * NT_ + tid;
    const long stride = (long)gridDim.x * NT_;

    float acc = 0.0f;

    const int  n4 = n >> 2;
    const v4f* D4 = (const v4f*)D;
    const v4i* T4 = (const v4i*)T;

    // main vectorized streaming loop (NT hint: read-once data, keep L2 clean)
    for (long i = gid; i < n4; i += stride) {
        v4f d = __builtin_nontemporal_load(&D4[i]);
        v4i t = __builtin_nontemporal_load(&T4[i]);
#pragma unroll
        for (int k = 0; k < 4; ++k) {
            float dv  = d[k];
            int   tv  = t[k];
            float pos = fminf(0.0f,  dv - 1.0f);   // target == +1
            float neg = fminf(0.0f, -1.0f - dv);   // target == -1
            float c   = (tv == 1) ? pos : ((tv == -1) ? neg : 0.0f);
            acc += c;
        }
    }
    // scalar tail (N = 2^25 so normally empty; kept for generality)
    for (long i = (long)n4 * 4 + gid; i < n; i += stride) {
        float dv  = D[i];
        int   tv  = T[i];
        float pos = fminf(0.0f,  dv - 1.0f);
        float neg = fminf(0.0f, -1.0f - dv);
        acc += (tv == 1) ? pos : ((tv == -1) ? neg : 0.0f);
    }

    // block tree reduction in LDS (cost negligible vs. memory traffic)
    sdata[tid] = acc;
    __syncthreads();
#pragma unroll
    for (int s = NT_ / 2; s > 0; s >>= 1) {
        if (tid < s) sdata[tid] += sdata[tid + s];
        __syncthreads();
    }
    if (tid == 0) partials[blockIdx.x] = sdata[0];
}

// ---------------------------------------------------------------------------
// Pass 2: single wave32 reduces NB partials; cross-lane reduction done with
// V_WMMA_F32_16X16X4_F32 (ones-matrix trick, exact in f32):
//   A: vgpr0 = acc (lanes 0-15 -> A[m][0], lanes 16-31 -> A[m][2]), vgpr1 = 0
//   B: all ones  =>  D[m][n] = acc[m] + acc[m+16]  (broadcast over n)
// Lane L then sums its 8 D VGPRs:
//   lanes 0-15  : S_lo = sum(acc[0..7])  + sum(acc[16..23])
//   lanes 16-31 : S_hi = sum(acc[8..15]) + sum(acc[24..31])
// One xor-16 shuffle + add completes the 32-lane sum.
// ---------------------------------------------------------------------------
__global__ __launch_bounds__(32) void gan_hinge_final(
    const float* __restrict__ partials, int nPartials,
    float* __restrict__ out, int nTotal)
{
    const int lane = threadIdx.x;

    float acc = 0.0f;
    const int  n4 = nPartials >> 2;
    const v4f* p4 = (const v4f*)partials;
    for (int i = lane; i < n4; i += 32) {
        v4f v = p4[i];
        acc += (v[0] + v[1]) + (v[2] + v[3]);
    }
    for (int i = n4 * 4 + lane; i < nPartials; i += 32) acc += partials[i];

    // WMMA cross-lane reduction (EXEC is all-1s: full single wave, no divergence)
    v2f a; a[0] = acc;  a[1] = 0.0f;   // A-matrix: data in K=0/K=2 slots
    v2f b; b[0] = 1.0f; b[1] = 1.0f;   // B-matrix: all ones
    v8f c = {};
    c = __builtin_amdgcn_wmma_f32_16x16x4_f32(
            /*neg_a=*/false, a, /*neg_b=*/false, b,
            /*c_mod=*/(short)0, c, /*reuse_a=*/false, /*reuse_b=*/false);

    float s = ((c[0] + c[1]) + (c[2] + c[3])) + ((c[4] + c[5]) + (c[6] + c[7]));
    float other = __shfl(s, lane ^ 16, 32);
    float total = s + other;

    if (lane == 0) out[0] = total / (float)nTotal;  // match reference: sum / N
}

// ---------------------------------------------------------------------------
extern "C" void kernel_launch(void* const* d_in, const int* in_sizes, int n_in,
                              void* d_out, int out_size, void* d_ws, size_t ws_size,
                              hipStream_t stream)
{
    const float* D = (const float*)d_in[0];
    const int*   T = (const int*)d_in[1];
    float*       out = (float*)d_out;
    float*       partials = (float*)d_ws;
    const int    n = in_sizes[0];

    int nb = NB;
    if (ws_size < (size_t)nb * sizeof(float)) {
        nb = (int)(ws_size / sizeof(float)) & ~31;  // keep multiple of 32
        if (nb < 32) nb = 32;
    }

    gan_hinge_partial<<<nb, NT_, 0, stream>>>(D, T, partials, n);
    gan_hinge_final<<<1, 32, 0, stream>>>(partials, nb, out, n);
}